// LogicLayer_82789789597761
// MI455X (gfx1250) — compile-verified
//
#include <hip/hip_runtime.h>
#include <hip/hip_bf16.h>
#include <math.h>

#define LL_IN   16384   // x columns (per reference)
#define LL_OUT  16384   // output columns (per reference)
#define TB      256     // threads per block, main kernel (8 waves on wave32)

typedef float v2f  __attribute__((ext_vector_type(2)));
typedef float v4f  __attribute__((ext_vector_type(4)));
typedef float v8f  __attribute__((ext_vector_type(8)));

// GCC-style int4 vector: the async-to-LDS builtin's pointee type per hipcc's
// diagnostic ("__vector_size__(4 * sizeof(int)) int").
typedef int vsi4 __attribute__((vector_size(16)));
typedef __attribute__((address_space(1))) vsi4 g_vsi4;   // global
typedef __attribute__((address_space(3))) vsi4 l_vsi4;   // LDS

// gate_g(a,b) = M[g][0] + M[g][1]*a + M[g][2]*b + M[g][3]*a*b  (exact)
__device__ const float GateM[16][4] = {
  { 0.f,  0.f,  0.f,  0.f},  // 0
  { 0.f,  0.f,  0.f,  1.f},  // ab
  { 0.f,  1.f,  0.f, -1.f},  // a - ab
  { 0.f,  1.f,  0.f,  0.f},  // a
  { 0.f,  0.f,  1.f, -1.f},  // b - ab
  { 0.f,  0.f,  1.f,  0.f},  // b
  { 0.f,  1.f,  1.f, -2.f},  // a + b - 2ab
  { 0.f,  1.f,  1.f, -1.f},  // a + b - ab
  { 1.f, -1.f, -1.f,  1.f},  // 1 - (a + b - ab)
  { 1.f, -1.f, -1.f,  2.f},  // 1 - (a + b - 2ab)
  { 1.f,  0.f, -1.f,  0.f},  // 1 - b
  { 1.f,  0.f, -1.f,  1.f},  // 1 - b + ab
  { 1.f, -1.f,  0.f,  0.f},  // 1 - a
  { 1.f, -1.f,  0.f,  1.f},  // 1 - a + ab
  { 1.f,  0.f,  0.f, -1.f},  // 1 - ab
  { 1.f,  0.f,  0.f,  0.f}}; // 1

// --------------------------------------------------------------------------
// Kernel 1: one wave per 16 output columns.
//   probs = softmax(weights[16x16])          (lanes 0-15, one row each)
//   P(16x4) = probs(16x16) @ GateM(16x4)     via 4x V_WMMA_F32_16X16X4_F32
// Also packs int64 idx_a/idx_b into an int2 table.
// A 16x4 layout: lanes 0-15 -> {K=0,K=1}, lanes 16-31 -> {K=2,K=3} (M = lane&15)
// D 16x16 layout: VGPR r holds M=r (lanes 0-15) / M=r+8 (lanes 16-31), N = lane&15
// --------------------------------------------------------------------------
__global__ __launch_bounds__(32) void ll_coef_kernel(
    const float* __restrict__ w, const long long* __restrict__ ia,
    const long long* __restrict__ ib, v4f* __restrict__ coef,
    int2* __restrict__ idxp) {
  __shared__ float probs[16][17];
  const int lane  = threadIdx.x;          // 0..31, single wave32
  const int nbase = blockIdx.x * 16;
  if (lane < 16) {
    const int n = nbase + lane;
    const float* wr = w + n * 16;
    float v[16];
    float mx = wr[0];
#pragma unroll
    for (int g = 0; g < 16; ++g) { v[g] = wr[g]; mx = fmaxf(mx, v[g]); }
    float s = 0.f;
#pragma unroll
    for (int g = 0; g < 16; ++g) { v[g] = expf(v[g] - mx); s += v[g]; }
    const float inv = 1.f / s;
#pragma unroll
    for (int g = 0; g < 16; ++g) probs[lane][g] = v[g] * inv;
    idxp[n] = make_int2((int)ia[n], (int)ib[n]);
  }
  __syncthreads();   // reconverge: EXEC must be all-1s for WMMA

  const int col = lane & 15;              // B/D column (N)
  const int m   = lane & 15;              // A row (M)
  const int kb  = (lane >> 4) << 1;       // K base: 0 (lanes 0-15) or 2 (16-31)
  v8f acc = {0.f,0.f,0.f,0.f,0.f,0.f,0.f,0.f};
#pragma unroll
  for (int c = 0; c < 4; ++c) {           // K = 16 gates as 4 chunks of 4
    v2f A, B;
    A.x = probs[m][4*c + kb];
    A.y = probs[m][4*c + kb + 1];
    B.x = (col < 4) ? GateM[4*c + kb][col]     : 0.f;  // GateM padded 4x16
    B.y = (col < 4) ? GateM[4*c + kb + 1][col] : 0.f;
    acc = __builtin_amdgcn_wmma_f32_16x16x4_f32(
        /*neg_a=*/false, A, /*neg_b=*/false, B,
        /*c_mod=*/(short)0, acc, /*reuse_a=*/false, /*reuse_b=*/false);
  }
  if (col < 4) {                          // only first 4 D columns are real
    const int rbase = (lane < 16) ? 0 : 8;
#pragma unroll
    for (int r = 0; r < 8; ++r)
      ((float*)&coef[nbase + rbase + r])[col] = acc[r];
  }
}

// --------------------------------------------------------------------------
// Kernel 2: one block per x-row. Stage the 64 KB row in LDS (async load-to-LDS
// when available), then each lane does banked LDS gathers + 3 FMAs per output,
// streaming the result with non-temporal stores.
// --------------------------------------------------------------------------
#if defined(__gfx1250__) && __has_builtin(__builtin_amdgcn_global_load_async_to_lds_b128)
#define LL_ASYNC 1
#endif

__global__ __launch_bounds__(TB) void ll_main_kernel(
    const float* __restrict__ x, const v4f* __restrict__ coef,
    const int2* __restrict__ idxp, float* __restrict__ out) {
  __shared__ __align__(16) float row[LL_IN];          // 64 KB of the 320 KB LDS
  const int    t = threadIdx.x;
  const size_t r = blockIdx.x;
  const v4f* __restrict__ src = (const v4f*)(x + r * (size_t)LL_IN);

#ifdef LL_ASYNC
  {
#pragma unroll
    for (int i = 0; i < LL_IN / 4 / TB; ++i) {        // 16 x b128 per lane
      const int e = t + i * TB;
      __builtin_amdgcn_global_load_async_to_lds_b128(
          (g_vsi4*)(uintptr_t)(src + e),
          (l_vsi4*)(uint32_t)(uintptr_t)(&row[4 * e]),
          /*offset=*/0, /*cpol=*/0);
    }
#if __has_builtin(__builtin_amdgcn_s_wait_asynccnt)
    __builtin_amdgcn_s_wait_asynccnt(0);
#else
    asm volatile("s_wait_asynccnt 0" ::: "memory");
#endif
  }
#else
#pragma unroll
  for (int i = 0; i < LL_IN / 4 / TB; ++i) {
    const int e = t + i * TB;
    ((v4f*)row)[e] = __builtin_nontemporal_load(src + e);  // row used once: NT
  }
#endif
  __syncthreads();

  float* __restrict__ orow = out + r * (size_t)LL_OUT;
#pragma unroll 4
  for (int j = 0; j < LL_OUT / TB; ++j) {
    const int  n  = t + j * TB;           // coalesced across lanes
    const int2 ii = idxp[n];              // L2-resident (reused 2048x)
    const float a = row[ii.x];            // LDS gather
    const float b = row[ii.y];            // LDS gather
    const v4f  cf = coef[n];              // L2-resident
    // P0 + P1*a + P2*b + P3*a*b  ==  a*(P1 + P3*b) + (P0 + P2*b)
    const float res = fmaf(a, fmaf(cf.w, b, cf.y), fmaf(cf.z, b, cf.x));
    __builtin_nontemporal_store(res, &orow[n]);       // 134 MB stream: NT
  }
}

// --------------------------------------------------------------------------
extern "C" void kernel_launch(void* const* d_in, const int* in_sizes, int n_in,
                              void* d_out, int out_size, void* d_ws, size_t ws_size,
                              hipStream_t stream) {
  const float*     x  = (const float*)d_in[0];      // (B, 16384) f32
  const float*     w  = (const float*)d_in[1];      // (16384, 16) f32
  const long long* ia = (const long long*)d_in[2];  // (16384,) i64
  const long long* ib = (const long long*)d_in[3];  // (16384,) i64
  float*           o  = (float*)d_out;              // (B, 16384) f32

  const int B = in_sizes[0] / LL_IN;                // 2048 for the reference

  // scratch layout: [0, 256KB) = float4 coef[16384]; [256KB, 384KB) = int2 idx[16384]
  v4f*  coef = (v4f*)d_ws;
  int2* idxp = (int2*)((char*)d_ws + (size_t)LL_OUT * sizeof(v4f));

  ll_coef_kernel<<<LL_OUT / 16, 32, 0, stream>>>(w, ia, ib, coef, idxp);
  ll_main_kernel<<<B, TB, 0, stream>>>(x, coef, idxp, o);
}